// Nais_66924180406355
// MI455X (gfx1250) — compile-verified
//
#include <hip/hip_runtime.h>
#include <math.h>

// MI455X / gfx1250, wave32. All matrix math via V_WMMA_F32_16X16X4_F32 (fp32,
// full reference precision — problem is memory/latency bound so no reason to
// quantize). Nonlinear relu-dot stage is fp32 VALU from LDS.

typedef __attribute__((ext_vector_type(2))) float v2f;
typedef __attribute__((ext_vector_type(8))) float v8f;

#define EMB   128
#define NW    128
#define NT    64
#define WIN   200
#define BATCH 128

// ---------------------------------------------------------------------------
// proj: H[b, r, n] = sum_e vecs[ids[b,r]][e] * W[n*wld + wo + e]  (+ bias[n])
// One wave per 16(M=r) x 16(N=n) tile, K=EMB=128 -> 32 wmma_f32_16x16x4 steps.
// blockDim = 256 (8 waves); wave w handles n-tile w; blockIdx -> (b, r-tile).
// ---------------------------------------------------------------------------
__global__ __launch_bounds__(256) void proj_kernel(
    const float* __restrict__ vecs, const int* __restrict__ ids,
    const float* __restrict__ W, int wld, int wo,
    const float* __restrict__ bias, float* __restrict__ H,
    int R, int RT, int Hld)
{
  const int nt   = threadIdx.x >> 5;          // 0..7 : n-tile
  const int lane = threadIdx.x & 31;
  const int l16  = lane & 15;
  const int kh   = lane >> 4;                 // K-half select (ISA A/B layout)
  const int rt   = blockIdx.x % RT;
  const int b    = blockIdx.x / RT;

  const int r   = rt * 16 + l16;              // A row (M)
  const int rc  = (r < R) ? r : (R - 1);      // clamp (padded rows unused)
  const int rid = ids[(size_t)b * R + rc];
  const float* __restrict__ arow = vecs + (size_t)rid * EMB;

  const int n = nt * 16 + l16;                // B col (N)
  const float* __restrict__ brow = W + (size_t)n * wld + wo;

  v8f acc = {0.f, 0.f, 0.f, 0.f, 0.f, 0.f, 0.f, 0.f};
#pragma unroll
  for (int kk = 0; kk < EMB / 4; ++kk) {
    const int e0 = kk * 4 + kh * 2;
    v2f a  = *(const v2f*)(arow + e0);        // A[M][K0], A[M][K0+1]
    v2f bb = *(const v2f*)(brow + e0);        // B[K0][N], B[K0+1][N]
    acc = __builtin_amdgcn_wmma_f32_16x16x4_f32(
        false, a, false, bb, (short)0, acc, false, false);
  }

  const float bv = bias ? bias[n] : 0.0f;
  const size_t rowbase = ((size_t)b * Hld + rt * 16) * NW;
#pragma unroll
  for (int v = 0; v < 8; ++v) {
    const int m = v + 8 * kh;                 // D layout: M = v + 8*(lane/16)
    H[rowbase + (size_t)m * NW + n] = acc[v] + bv;
  }
}

// ---------------------------------------------------------------------------
// attn: per (b, 16-row t-tile):
//   att[t,w] = sum_n relu(hpb[t,n] + hq[w,n]) * h_w[n]   (W_b folded in hpb,
//   h_b dropped: softmax is shift-invariant), mask -> -inf, softmax over w,
//   out[t,e] = (sum_w att[t,w] * cvecs[cit[w]][e]) * rsqrt(1000 - cnt).
// Final GEMM: fp32 WMMA, M=16 (t) x N=128 (e, 8 waves) x K=208 (w, padded 0).
// ---------------------------------------------------------------------------
#define WC 64   // hq chunk rows staged in LDS

__global__ __launch_bounds__(256) void attn_kernel(
    const float* __restrict__ hp, const float* __restrict__ hq,
    const float* __restrict__ cvecs, const int* __restrict__ citems,
    const unsigned char* __restrict__ mask, const float* __restrict__ h_w,
    float* __restrict__ out)
{
  __shared__ float s_hp[16][132];     // +4 pad: conflict-free, 16B aligned rows
  __shared__ float s_hq[WC][132];
  __shared__ float s_att[16][212];    // 208 cols used (200 + zero pad)
  __shared__ float s_hw[NW];
  __shared__ int   s_cit[208];
  __shared__ int   s_cnt;

  const int tid = threadIdx.x;
  const int b   = blockIdx.x >> 2;
  const int t0  = (blockIdx.x & 3) * 16;

  if (tid == 0) s_cnt = 0;
  if (tid < NW) s_hw[tid] = h_w[tid];
  for (int i = tid; i < 16 * NW; i += 256) {
    const int r = i >> 7, n = i & 127;
    s_hp[r][n] = hp[((size_t)b * NT + t0 + r) * NW + n];
  }
  for (int w = tid; w < 208; w += 256)
    s_cit[w] = citems[(size_t)b * WIN + (w < WIN ? w : WIN - 1)];
  __syncthreads();
  if (tid < WIN) atomicAdd(&s_cnt, (int)mask[(size_t)b * WIN + tid]);

  // ---- att = relu-dot, streaming hq through LDS in 64-row chunks ----
  for (int base = 0; base < WIN; base += WC) {
    const int wc = (WIN - base < WC) ? (WIN - base) : WC;
    __syncthreads();
    for (int i = tid; i < wc * NW; i += 256) {
      const int r = i >> 7, n = i & 127;
      s_hq[r][n] = hq[((size_t)b * 208 + base + r) * NW + n];
    }
    __syncthreads();
    for (int idx = tid; idx < 16 * wc; idx += 256) {
      const int t = idx & 15, wl = idx >> 4, w = base + wl;
      const float4* __restrict__ p4 = (const float4*)s_hp[t];
      const float4* __restrict__ q4 = (const float4*)s_hq[wl];
      const float4* __restrict__ h4 = (const float4*)s_hw;
      float acc = 0.f;
#pragma unroll
      for (int j = 0; j < NW / 4; ++j) {
        const float4 p = p4[j], q = q4[j], h = h4[j];
        acc = fmaf(fmaxf(p.x + q.x, 0.f), h.x, acc);
        acc = fmaf(fmaxf(p.y + q.y, 0.f), h.y, acc);
        acc = fmaf(fmaxf(p.z + q.z, 0.f), h.z, acc);
        acc = fmaf(fmaxf(p.w + q.w, 0.f), h.w, acc);
      }
      s_att[t][w] = mask[(size_t)b * WIN + w] ? -INFINITY : acc;
    }
  }
  __syncthreads();

  // ---- masked softmax over w (one wave per 2 rows, shuffle reductions) ----
  const int wid = tid >> 5, lane = tid & 31;
  for (int t = wid; t < 16; t += 8) {
    float m = -INFINITY;
    for (int w = lane; w < WIN; w += 32) m = fmaxf(m, s_att[t][w]);
#pragma unroll
    for (int off = 16; off > 0; off >>= 1) m = fmaxf(m, __shfl_xor(m, off, 32));
    float s = 0.f;
    for (int w = lane; w < WIN; w += 32) {
      const float e = expf(s_att[t][w] - m);
      s_att[t][w] = e;
      s += e;
    }
#pragma unroll
    for (int off = 16; off > 0; off >>= 1) s += __shfl_xor(s, off, 32);
    const float inv = 1.0f / s;
    for (int w = lane; w < WIN; w += 32) s_att[t][w] *= inv;
    if (lane < 8) s_att[t][WIN + lane] = 0.f;   // zero K-pad 200..207
  }
  __syncthreads();

  // ---- out tile: 16x16 per wave, K=208 via 52 fp32 WMMAs ----
  const float inv_norm = rsqrtf(1000.0f - (float)s_cnt);
  const int nt = wid, l16 = lane & 15, kh = lane >> 4;
  const int e = nt * 16 + l16;
  v8f acc = {0.f, 0.f, 0.f, 0.f, 0.f, 0.f, 0.f, 0.f};
#pragma unroll 4
  for (int kk = 0; kk < 52; ++kk) {
    const int K0 = kk * 4 + kh * 2;
    v2f a = *(const v2f*)&s_att[l16][K0];      // A[M=t][K0], [K0+1]
    v2f bb;
    bb.x = cvecs[(size_t)s_cit[K0] * EMB + e]; // B[K0][N=e]
    bb.y = cvecs[(size_t)s_cit[K0 + 1] * EMB + e];
    acc = __builtin_amdgcn_wmma_f32_16x16x4_f32(
        false, a, false, bb, (short)0, acc, false, false);
  }
#pragma unroll
  for (int v = 0; v < 8; ++v) {
    const int t = t0 + v + 8 * kh;
    out[((size_t)b * NT + t) * EMB + e] = acc[v] * inv_norm;
  }
}

// ---------------------------------------------------------------------------
extern "C" void kernel_launch(void* const* d_in, const int* in_sizes, int n_in,
                              void* d_out, int out_size, void* d_ws, size_t ws_size,
                              hipStream_t stream) {
  const float* tvecs = (const float*)d_in[0];   // (100000,128) f32
  const float* cvecs = (const float*)d_in[1];   // (100000,128) f32
  const float* W_w   = (const float*)d_in[2];   // (128,256)    f32
  const float* W_b   = (const float*)d_in[3];   // (128,)       f32
  const float* h_w   = (const float*)d_in[4];   // (1,128)      f32
  // d_in[5] = h_b : dropped (softmax shift-invariance)
  const int* titems  = (const int*)d_in[6];     // (128,64)  i32
  const int* citems  = (const int*)d_in[7];     // (128,200) i32
  const unsigned char* mask = (const unsigned char*)d_in[8]; // (128,200) bool
  float* out = (float*)d_out;                   // (128,64,128) f32

  float* hp = (float*)d_ws;                         // B*64*128  = 4 MB
  float* hq = hp + (size_t)BATCH * NT * NW;         // B*208*128 = 13.6 MB

  // hp = q . Wp^T + W_b   (Wp = W_w[:, :128])
  proj_kernel<<<BATCH * 4, 256, 0, stream>>>(
      tvecs, titems, W_w, 2 * EMB, 0, W_b, hp, NT, 4, NT);
  // hq = k . Wq^T         (Wq = W_w[:, 128:])
  proj_kernel<<<BATCH * 13, 256, 0, stream>>>(
      cvecs, citems, W_w, 2 * EMB, EMB, nullptr, hq, WIN, 13, 208);
  // fused relu-dot + softmax + out GEMM + norm
  attn_kernel<<<BATCH * 4, 256, 0, stream>>>(
      hp, hq, cvecs, citems, mask, h_w, out);
}